// DLinear_38268158608117
// MI455X (gfx1250) — compile-verified
//
#include <hip/hip_runtime.h>

// D = A x B + C fragments for V_WMMA_F32_16X16X4_F32 (wave32)
typedef __attribute__((ext_vector_type(2))) float v2f;  // A/B fragment: 2 VGPRs
typedef __attribute__((ext_vector_type(8))) float v8f;  // C/D fragment: 8 VGPRs

#define B_TOTAL 4096
#define T_DIM   24     // K of the GEMM
#define F_DIM   256
#define O_DIM   24     // N of the GEMM
#define TGT     2      // only f = 0,1 survive the output slice

// One wave handles a 16-batch tile for BOTH surviving features.
//   A[m][k] = x[b0+m, t=kstep+k, f]        (16x4 per step, f32)
//   B[k][n] = W[f, o=n(+16), t=kstep+k]    (4x16 per step, f32; o>=24 zero-padded)
//   C[m][n] = out[b0+m, o=n, f]  (+ bias[f,o])
__global__ __launch_bounds__(256) void dlinear_wmma_f32(
    const float* __restrict__ x,    // (4096, 24, 256)
    const float* __restrict__ W,    // (256, 24, 24)
    const float* __restrict__ bias, // (256, 24)
    float* __restrict__ out)        // (4096, 24, 2)
{
  const int lane = threadIdx.x & 31;
  const int wave = threadIdx.x >> 5;
  const int tile = blockIdx.x * 8 + wave;   // 256 tiles of 16 batches
  const int b0   = tile * 16;

  const int  mrow = lane & 15;  // A row / B,C column within the lane group
  const int  grp  = lane >> 4;  // selects K pair (A/B) and M+8 (C)

  // Second N-tile covers o = 16..23; lanes with mrow>=8 are padding.
  const int   oHi   = 16 + mrow;
  const bool  oHiOk = (oHi < O_DIM);
  const int   oHiC  = oHiOk ? oHi : 0;        // clamped index (uniform flow, no EXEC change)
  const float oHiM  = oHiOk ? 1.0f : 0.0f;    // zero-mask for padded B rows

  v8f acc00 = {};  // f=0, o 0..15
  v8f acc01 = {};  // f=0, o 16..23 (padded)
  v8f acc10 = {};  // f=1, o 0..15
  v8f acc11 = {};  // f=1, o 16..23 (padded)

  const float* xrow = x + (size_t)(b0 + mrow) * (T_DIM * F_DIM);
  const float* w0   = W;                    // W[0] : 24x24
  const float* w1   = W + O_DIM * T_DIM;    // W[1] : 24x24

#pragma unroll
  for (int k = 0; k < T_DIM; k += 4) {
    const int t0 = k + 2 * grp;             // this lane group's K pair

    // x gather: f=0 and f=1 are adjacent -> one b64 load per t, row touched once.
    v2f p0 = *(const v2f*)(xrow + (size_t)t0 * F_DIM);
    v2f p1 = *(const v2f*)(xrow + (size_t)(t0 + 1) * F_DIM);
    v2f a_f0 = { p0.x, p1.x };              // A fragment, feature 0
    v2f a_f1 = { p0.y, p1.y };              // A fragment, feature 1

    // W fragments: consecutive t -> b64 loads; 4.6 KB total, L2-resident.
    v2f b00 = *(const v2f*)(w0 + mrow * T_DIM + t0);
    v2f b10 = *(const v2f*)(w1 + mrow * T_DIM + t0);
    v2f b01 = *(const v2f*)(w0 + oHiC * T_DIM + t0) * oHiM;  // branchless pad
    v2f b11 = *(const v2f*)(w1 + oHiC * T_DIM + t0) * oHiM;

    // EXEC is all-1s here (no divergence above) — WMMA requirement.
    acc00 = __builtin_amdgcn_wmma_f32_16x16x4_f32(false, a_f0, false, b00, (short)0, acc00, false, false);
    acc01 = __builtin_amdgcn_wmma_f32_16x16x4_f32(false, a_f0, false, b01, (short)0, acc01, false, false);
    acc10 = __builtin_amdgcn_wmma_f32_16x16x4_f32(false, a_f1, false, b10, (short)0, acc10, false, false);
    acc11 = __builtin_amdgcn_wmma_f32_16x16x4_f32(false, a_f1, false, b11, (short)0, acc11, false, false);
  }

  // Bias: depends only on the C column (o) of this lane.
  const float bs00 = bias[0 * O_DIM + mrow];
  const float bs10 = bias[1 * O_DIM + mrow];
  const float bs01 = bias[0 * O_DIM + oHiC] * oHiM;
  const float bs11 = bias[1 * O_DIM + oHiC] * oHiM;

  // Stores: C VGPR r holds M = r + 8*grp, N = lane&15. Pack (f0,f1) -> b64 store.
#pragma unroll
  for (int r = 0; r < 8; ++r) {
    const int bb = b0 + r + 8 * grp;
    v2f o0;
    o0.x = acc00[r] + bs00;
    o0.y = acc10[r] + bs10;
    *(v2f*)(out + (size_t)bb * (O_DIM * TGT) + mrow * TGT) = o0;
  }
  if (oHiOk) {  // divergent only here, after all WMMAs
#pragma unroll
    for (int r = 0; r < 8; ++r) {
      const int bb = b0 + r + 8 * grp;
      v2f o1;
      o1.x = acc01[r] + bs01;
      o1.y = acc11[r] + bs11;
      *(v2f*)(out + (size_t)bb * (O_DIM * TGT) + oHi * TGT) = o1;
    }
  }
}

extern "C" void kernel_launch(void* const* d_in, const int* in_sizes, int n_in,
                              void* d_out, int out_size, void* d_ws, size_t ws_size,
                              hipStream_t stream) {
  const float* x    = (const float*)d_in[0];  // (4096, 24, 256) f32
  const float* W    = (const float*)d_in[1];  // (256, 24, 24)   f32
  const float* bias = (const float*)d_in[2];  // (256, 24)       f32
  float* out = (float*)d_out;                 // (4096, 24, 2)   f32

  // 256 batch tiles, 8 waves (one tile each) per 256-thread block -> 32 blocks.
  dim3 grid(B_TOTAL / 16 / 8);
  dim3 block(256);
  dlinear_wmma_f32<<<grid, block, 0, stream>>>(x, W, bias, out);
}